// HamiltonProductModule_32401233281622
// MI455X (gfx1250) — compile-verified
//
#include <hip/hip_runtime.h>

// Quaternion Hamilton product, fp32, pure streaming.
// Roofline: 28 FLOP / 48 B per quaternion = 0.58 FLOP/byte -> HBM-bound
// (~17.3 us floor at 23.3 TB/s for 403 MB). Optimal path on MI455X:
//   - one quaternion == one b128 per lane, perfectly coalesced in wave32
//   - th:TH_LOAD_NT / TH_STORE_NT (403 MB touch-once > 192 MB L2)
//   - 4x unroll -> 8 outstanding global_load_b128 per thread
//   - uniform full-tile fast path: no exec masking, no VGPR zero-init
// WMMA deliberately not used: per-quaternion 4x4 matrices are all distinct,
// so any 16x16 WMMA tiling wastes 15/16 lanes and *increases* bytes moved.

typedef __attribute__((ext_vector_type(4))) float v4f;

__device__ __forceinline__ v4f hamilton(v4f a, v4f b) {
    // a = (w1,x1,y1,z1), b = (w2,x2,y2,z2) packed as .x,.y,.z,.w
    v4f r;
    r.x = a.x * b.x - a.y * b.y - a.z * b.z - a.w * b.w;  // out_w
    r.y = a.x * b.y + a.y * b.x + a.z * b.w - a.w * b.z;  // out_x
    r.z = a.x * b.z - a.y * b.w + a.z * b.x + a.w * b.y;  // out_y
    r.w = a.x * b.w + a.y * b.z - a.z * b.y + a.w * b.x;  // out_z
    return r;
}

#define HP_THREADS 256
#define HP_UNROLL 4
#define HP_TILE (HP_THREADS * HP_UNROLL)   // 1024 quaternions per block

__global__ __launch_bounds__(HP_THREADS) void hamilton_product_kernel(
    const v4f* __restrict__ q1, const v4f* __restrict__ q2,
    v4f* __restrict__ out, int nq)
{
    const int tile = blockIdx.x * HP_TILE;

    if (tile + HP_TILE <= nq) {
        // ---- full tile: straight-line, no guards, no exec masking ----
        const int i0 = tile + (int)threadIdx.x;
        v4f a[HP_UNROLL], b[HP_UNROLL];
        // Issue all 8 b128 NT loads back-to-back (LOADcnt soaks latency).
#pragma unroll
        for (int k = 0; k < HP_UNROLL; ++k) {
            a[k] = __builtin_nontemporal_load(&q1[i0 + k * HP_THREADS]);
            b[k] = __builtin_nontemporal_load(&q2[i0 + k * HP_THREADS]);
        }
#pragma unroll
        for (int k = 0; k < HP_UNROLL; ++k) {
            __builtin_nontemporal_store(hamilton(a[k], b[k]),
                                        &out[i0 + k * HP_THREADS]);
        }
    } else {
        // ---- tail tile (at most one block): guarded per element ----
#pragma unroll
        for (int k = 0; k < HP_UNROLL; ++k) {
            const int i = tile + (int)threadIdx.x + k * HP_THREADS;
            if (i < nq) {
                v4f a = __builtin_nontemporal_load(&q1[i]);
                v4f b = __builtin_nontemporal_load(&q2[i]);
                __builtin_nontemporal_store(hamilton(a, b), &out[i]);
            }
        }
    }
}

extern "C" void kernel_launch(void* const* d_in, const int* in_sizes, int n_in,
                              void* d_out, int out_size, void* d_ws, size_t ws_size,
                              hipStream_t stream) {
    const v4f* q1 = (const v4f*)d_in[0];
    const v4f* q2 = (const v4f*)d_in[1];
    v4f* out = (v4f*)d_out;

    const int nq = in_sizes[0] / 4;                // quaternion count
    const int grid = (nq + HP_TILE - 1) / HP_TILE; // 8192 for the ref shape

    hamilton_product_kernel<<<grid, HP_THREADS, 0, stream>>>(q1, q2, out, nq);
}